// Net_43490838839931
// MI455X (gfx1250) — compile-verified
//
#include <hip/hip_runtime.h>

// CDNA5 / gfx1250 neural-ODE RK4 kernel.
//
// Each wave32 integrates 32 batch elements (one per lane) for 100 time steps.
// Layer 1 of the 2->4->1 MLP runs on the matrix pipe via V_WMMA_F32_16X16X4_F32
// in a dual-group transposed formulation:
//   - B matrix (4x16) = the raw per-lane state registers {s, u}:
//       VGPR0 = {K0: s lanes 0-15 | K2: s lanes 16-31}
//       VGPR1 = {K1: u lanes 0-15 | K3: u lanes 16-31}
//   - A matrix (16x4) = W1^T for lane-group A in rows 0-3 (K0/K1) and W1^T for
//     lane-group B in rows 8-11 (K2/K3). Since D row r -> VGPR r lanes 0-15 and
//     row r+8 -> VGPR r lanes 16-31, the hidden activations for ALL 32 elements
//     land per-lane in D VGPRs 0-3.
//   - C matrix = b1[r] broadcast to all lanes of VGPR r (rows r and r+8).
// Layer 2 (h @ W2 + b2) reduces across registers, not lanes: 4 FMAs with
// SGPR-uniform weights. LeakyReLU = max(h, 0.01h), exact for slope < 1.

typedef __attribute__((ext_vector_type(2))) float v2f;
typedef __attribute__((ext_vector_type(8))) float v8f;

#define LEAKY_SLOPE 0.01f

struct Net {
    v2f a1;                          // layer-1 A-matrix (constant all loop)
    v8f c1;                          // layer-1 bias accumulator (constant)
    float w0, w1, w2, w3, bias2;     // layer-2 weights/bias (uniform -> SGPRs)
};

// dyn(s) = LeakyReLU([s,u] @ W1 + b1) @ W2 + b2, for all 32 lanes at once.
__device__ __forceinline__ float mlp_dyn(float s, float uu, const Net& net) {
    v2f b;
    b[0] = s;   // K0 row (lanes 0-15) and K2 row (lanes 16-31)
    b[1] = uu;  // K1 row (lanes 0-15) and K3 row (lanes 16-31)
    // h[j] (j=0..3) lands per-lane in d[0..3] for all 32 elements.
    v8f h = __builtin_amdgcn_wmma_f32_16x16x4_f32(
        false, net.a1, false, b, (short)0, net.c1, false, false);

    float l0 = fmaxf(h[0], LEAKY_SLOPE * h[0]);
    float l1 = fmaxf(h[1], LEAKY_SLOPE * h[1]);
    float l2 = fmaxf(h[2], LEAKY_SLOPE * h[2]);
    float l3 = fmaxf(h[3], LEAKY_SLOPE * h[3]);

    // Layer 2: per-lane dot across the four hidden registers.
    float o = fmaf(net.w0, l0, net.bias2);
    o = fmaf(net.w1, l1, o);
    o = fmaf(net.w2, l2, o);
    o = fmaf(net.w3, l3, o);
    return o;
}

__device__ __forceinline__ float rk4_step(float s, float uu, const Net& net) {
    float k1 = mlp_dyn(s, uu, net);
    float k2 = mlp_dyn(fmaf(0.5f, k1, s), uu, net);
    float k3 = mlp_dyn(fmaf(0.5f, k2, s), uu, net);
    float k4 = mlp_dyn(s + k3, uu, net);
    return fmaf(1.0f / 6.0f, k1 + 2.0f * k2 + 2.0f * k3 + k4, s);
}

__global__ __launch_bounds__(256) void node_rk4_wmma(
    const float* __restrict__ x, const float* __restrict__ u,
    const float* __restrict__ W1, const float* __restrict__ b1,
    const float* __restrict__ W2, const float* __restrict__ b2,
    float* __restrict__ out, int B)
{
    const int lane = threadIdx.x & 31;
    const long e = (long)blockIdx.x * blockDim.x + threadIdx.x;  // 1 elem/lane
    const long eload = (e < B) ? e : (long)(B - 1);
    const bool valid = (e < B);

    Net net;
    // A1 (16x4), A-layout: lanes 0-15 hold K0/K1 of row=lane,
    //                      lanes 16-31 hold K2/K3 of row=lane-16.
    // Group A: rows 0-3, K0/K1 = W1[:, j]      -> lanes 0-3
    // Group B: rows 8-11, K2/K3 = W1[:, j]     -> lanes 24-27
    net.a1[0] = (lane < 4) ? W1[lane]
              : ((lane >= 24 && lane < 28) ? W1[lane - 24] : 0.0f);
    net.a1[1] = (lane < 4) ? W1[4 + lane]
              : ((lane >= 24 && lane < 28) ? W1[4 + (lane - 24)] : 0.0f);
    // C1: VGPR r covers D rows r (lanes 0-15) and r+8 (lanes 16-31); both get
    // b1[r] for r=0..3, so broadcast to all lanes. Rows 4-7 / 12-15 stay 0.
    net.c1 = v8f{};
    net.c1[0] = b1[0];
    net.c1[1] = b1[1];
    net.c1[2] = b1[2];
    net.c1[3] = b1[3];
    // Layer-2 weights: uniform scalar loads -> SGPR operands for the FMAs.
    net.w0 = W2[0]; net.w1 = W2[1]; net.w2 = W2[2]; net.w3 = W2[3];
    net.bias2 = b2[0];

    const float uu = u[eload];
    float s = x[eload];

    float* optr = out + eload * 100;

    // 100 trajectory values (t=0 is x), stored as float4 every 4 steps.
    // 25 blocks x 4 rk4 calls = 100 steps; the last step is unused slack.
#pragma unroll 1
    for (int blk = 0; blk < 25; ++blk) {
        float4 v;
        v.x = s;
        s = rk4_step(s, uu, net); v.y = s;
        s = rk4_step(s, uu, net); v.z = s;
        s = rk4_step(s, uu, net); v.w = s;
        if (valid) {
            *reinterpret_cast<float4*>(optr + blk * 4) = v;  // 400 % 16 == 0
        }
        s = rk4_step(s, uu, net);  // advance to first value of next block
    }
}

extern "C" void kernel_launch(void* const* d_in, const int* in_sizes, int n_in,
                              void* d_out, int out_size, void* d_ws, size_t ws_size,
                              hipStream_t stream) {
    const float* x  = (const float*)d_in[0];
    const float* u  = (const float*)d_in[1];
    const float* W1 = (const float*)d_in[2];
    const float* b1 = (const float*)d_in[3];
    const float* W2 = (const float*)d_in[4];
    const float* b2 = (const float*)d_in[5];
    float* out = (float*)d_out;

    const int B = in_sizes[0];                      // 131072
    const int threads = 256;                        // 8 wave32 per block
    const int blocks = (B + threads - 1) / threads; // 512 blocks, 4096 waves

    node_rk4_wmma<<<blocks, threads, 0, stream>>>(x, u, W1, b1, W2, b2, out, B);
}